// Attention_18640158064834
// MI455X (gfx1250) — compile-verified
//
#include <hip/hip_runtime.h>
#include <hip/hip_bf16.h>
#include <math.h>

typedef float v2f __attribute__((ext_vector_type(2)));
typedef float v8f __attribute__((ext_vector_type(8)));

#define HEADS 8
#define DH    8
#define EMB   64

// ---- order-preserving float <-> uint encoding (for integer atomicMax) ----
__device__ __forceinline__ unsigned f2ord(float x) {
  unsigned b = __float_as_uint(x);
  return (b & 0x80000000u) ? ~b : (b | 0x80000000u);
}
__device__ __forceinline__ float ord2f(unsigned u) {
  unsigned b = (u & 0x80000000u) ? (u ^ 0x80000000u) : ~u;
  return __uint_as_float(b);
}

// -------------------------------------------------------------------------
// Out(rows,64) = A(rows,64) @ W(64,64) + bias(64), fp32 WMMA.
// One wave per 16x16 output tile; 4 waves/block cover the N=64 dimension.
// K=64 -> 16 chained V_WMMA_F32_16X16X4_F32 steps.
// -------------------------------------------------------------------------
__global__ __launch_bounds__(128) void gemm64_wmma(
    const float* __restrict__ A, const float* __restrict__ W,
    const float* __restrict__ bias, float* __restrict__ Out, int rows)
{
  const int lane = threadIdx.x & 31;
  const int wv   = threadIdx.x >> 5;      // 0..3 -> N tile
  const int row0 = blockIdx.x << 4;
  const int n0   = wv << 4;
  const int half = lane >> 4;             // 0 | 1
  const int l15  = lane & 15;
  if (row0 >= rows) return;               // uniform per block (never splits EXEC)

  // C/D layout: vgpr i -> M = i + 8*half, N = l15. Seed with bias broadcast.
  v8f c;
  const float bb = bias[n0 + l15];
#pragma unroll
  for (int i = 0; i < 8; ++i) c[i] = bb;

  const float* arow = A + (size_t)(row0 + l15) * EMB;
#pragma unroll
  for (int kk = 0; kk < 16; ++kk) {
    const int k0 = (kk << 2) + (half << 1);
    // A 16x4: lane holds A[M=l15][K=k0+j] in vgpr j
    v2f a = *(const v2f*)(arow + k0);
    // B 4x16: lane holds W[K=k0+j][N=n0+l15] in vgpr j
    v2f b;
    b.x = W[(k0 + 0) * EMB + n0 + l15];
    b.y = W[(k0 + 1) * EMB + n0 + l15];
    c = __builtin_amdgcn_wmma_f32_16x16x4_f32(false, a, false, b, (short)0, c,
                                              false, false);
  }
#pragma unroll
  for (int i = 0; i < 8; ++i)
    Out[(size_t)(row0 + (half << 3) + i) * EMB + n0 + l15] = c[i];
}

// -------------------------------------------------------------------------
// Init: seg_max = encode(-inf), seg_sum = 0, accum = 0
// -------------------------------------------------------------------------
__global__ __launch_bounds__(256) void init_kernel(
    unsigned* __restrict__ seg_max, float* __restrict__ seg_sum,
    float* __restrict__ accum, int nEH, int nE64)
{
  int t = blockIdx.x * blockDim.x + threadIdx.x;
  if (t < nEH) { seg_max[t] = 0x007FFFFFu; /* = f2ord(-inf) */ seg_sum[t] = 0.0f; }
  if (t < nE64) accum[t] = 0.0f;
}

// -------------------------------------------------------------------------
// Pass A: per (pair, head) thread -> logit, atomicMax into seg_max[dst,h].
// Consecutive threads cover heads of one pair, so the 256B gathered rows
// coalesce into b128 loads (L2-resident: q/k/v = 154MB < 192MB L2).
// -------------------------------------------------------------------------
__global__ __launch_bounds__(256) void seg_max_kernel(
    const float* __restrict__ q, const float* __restrict__ k,
    const int* __restrict__ src, const int* __restrict__ dst,
    const float* __restrict__ bias, unsigned* __restrict__ seg_max, int Mtot)
{
  int t = blockIdx.x * blockDim.x + threadIdx.x;
  if (t >= Mtot * HEADS) return;
  const int m = t >> 3, h = t & 7;
  const int s = src[m], d = dst[m];
  const float4* qp = (const float4*)(q + (size_t)d * EMB + h * DH);
  const float4* kp = (const float4*)(k + (size_t)s * EMB + h * DH);
  float4 q0 = qp[0], q1 = qp[1];
  float4 k0 = kp[0], k1 = kp[1];
  float dot = q0.x * k0.x + q0.y * k0.y + q0.z * k0.z + q0.w * k0.w
            + q1.x * k1.x + q1.y * k1.y + q1.z * k1.z + q1.w * k1.w;
  float logit = dot * 0.35355339059327378f + bias[t];   // D^-0.5, D=8
  atomicMax(&seg_max[d * HEADS + h], f2ord(logit));
}

// -------------------------------------------------------------------------
// Pass B: recompute logit, ex = exp(logit - segmax); fuse segment-sum of ex
// and weighted segment-sum of ex*v[src] via f32 atomics (L2 atomic units).
// -------------------------------------------------------------------------
__global__ __launch_bounds__(256) void attn_accum_kernel(
    const float* __restrict__ q, const float* __restrict__ k,
    const float* __restrict__ v,
    const int* __restrict__ src, const int* __restrict__ dst,
    const float* __restrict__ bias, const unsigned* __restrict__ seg_max,
    float* __restrict__ seg_sum, float* __restrict__ accum, int Mtot)
{
  int t = blockIdx.x * blockDim.x + threadIdx.x;
  if (t >= Mtot * HEADS) return;
  const int m = t >> 3, h = t & 7;
  const int s = src[m], d = dst[m];
  const float4* qp = (const float4*)(q + (size_t)d * EMB + h * DH);
  const float4* kp = (const float4*)(k + (size_t)s * EMB + h * DH);
  float4 q0 = qp[0], q1 = qp[1];
  float4 k0 = kp[0], k1 = kp[1];
  float dot = q0.x * k0.x + q0.y * k0.y + q0.z * k0.z + q0.w * k0.w
            + q1.x * k1.x + q1.y * k1.y + q1.z * k1.z + q1.w * k1.w;
  float logit = dot * 0.35355339059327378f + bias[t];

  float mx = ord2f(seg_max[d * HEADS + h]);
  if (mx < -1e30f) mx = 0.0f;            // reference: where(isfinite, max, 0)
  float ex = __expf(logit - mx);

  atomicAdd(&seg_sum[d * HEADS + h], ex);
  const float4* vp = (const float4*)(v + (size_t)s * EMB + h * DH);
  float4 v0 = vp[0], v1 = vp[1];
  float* op = accum + (size_t)d * EMB + h * DH;
  atomicAdd(op + 0, ex * v0.x);
  atomicAdd(op + 1, ex * v0.y);
  atomicAdd(op + 2, ex * v0.z);
  atomicAdd(op + 3, ex * v0.w);
  atomicAdd(op + 4, ex * v1.x);
  atomicAdd(op + 5, ex * v1.y);
  atomicAdd(op + 6, ex * v1.z);
  atomicAdd(op + 7, ex * v1.w);
}

// -------------------------------------------------------------------------
// Normalize: accum[e,h,:] /= (seg_sum[e,h] + 1e-16)
// -------------------------------------------------------------------------
__global__ __launch_bounds__(256) void normalize_kernel(
    float* __restrict__ accum, const float* __restrict__ seg_sum, int nEH)
{
  int t = blockIdx.x * blockDim.x + threadIdx.x;
  if (t >= nEH) return;
  const int e = t >> 3, h = t & 7;
  const float inv = 1.0f / (seg_sum[t] + 1e-16f);
  float* p = accum + (size_t)e * EMB + h * DH;
#pragma unroll
  for (int i = 0; i < 8; ++i) p[i] *= inv;
}

extern "C" void kernel_launch(void* const* d_in, const int* in_sizes, int n_in,
                              void* d_out, int out_size, void* d_ws, size_t ws_size,
                              hipStream_t stream) {
  const float* feat = (const float*)d_in[0];   // (E, 64)
  const int*   e2e  = (const int*)d_in[1];     // (2, M)
  const float* abia = (const float*)d_in[2];   // (M, 8)
  const float* Wq = (const float*)d_in[3]; const float* bq = (const float*)d_in[4];
  const float* Wk = (const float*)d_in[5]; const float* bk = (const float*)d_in[6];
  const float* Wv = (const float*)d_in[7]; const float* bv = (const float*)d_in[8];
  const float* Wo = (const float*)d_in[9]; const float* bo = (const float*)d_in[10];

  const int E = in_sizes[0] / EMB;     // 200000 (multiple of 16)
  const int M = in_sizes[1] / 2;       // 3200000
  const int* src = e2e;
  const int* dst = e2e + M;

  // workspace carve-up (fp32): q,k,v,accum (E*64 each) + seg_sum,seg_max (E*8)
  float* ws      = (float*)d_ws;
  float* q       = ws;
  float* kbuf    = q     + (size_t)E * EMB;
  float* vbuf    = kbuf  + (size_t)E * EMB;
  float* accum   = vbuf  + (size_t)E * EMB;
  float* seg_sum = accum + (size_t)E * EMB;
  unsigned* seg_max = (unsigned*)(seg_sum + (size_t)E * HEADS);

  const int nEH  = E * HEADS;
  const int nE64 = E * EMB;
  init_kernel<<<(nE64 + 255) / 256, 256, 0, stream>>>(seg_max, seg_sum, accum,
                                                      nEH, nE64);

  const int gemmBlocks = (E + 15) / 16;
  gemm64_wmma<<<gemmBlocks, 128, 0, stream>>>(feat, Wq, bq, q, E);
  gemm64_wmma<<<gemmBlocks, 128, 0, stream>>>(feat, Wk, bk, kbuf, E);
  gemm64_wmma<<<gemmBlocks, 128, 0, stream>>>(feat, Wv, bv, vbuf, E);

  const long pairThreads = (long)M * HEADS;
  const int  pairBlocks  = (int)((pairThreads + 255) / 256);
  seg_max_kernel<<<pairBlocks, 256, 0, stream>>>(q, kbuf, src, dst, abia,
                                                 seg_max, M);
  attn_accum_kernel<<<pairBlocks, 256, 0, stream>>>(q, kbuf, vbuf, src, dst,
                                                    abia, seg_max, seg_sum,
                                                    accum, M);
  normalize_kernel<<<(nEH + 255) / 256, 256, 0, stream>>>(accum, seg_sum, nEH);

  gemm64_wmma<<<gemmBlocks, 128, 0, stream>>>(accum, Wo, bo, (float*)d_out, E);
}